// AdaptiveINR_58274116272327
// MI455X (gfx1250) — compile-verified
//
#include <hip/hip_runtime.h>
#include <stdint.h>

#define IMG_H 1024
#define IMG_W 1024
#define NB    8
#define TH    512
#define TW    512

typedef __attribute__((ext_vector_type(2))) float v2f;
typedef __attribute__((ext_vector_type(8))) float v8f;

// ---------------------------------------------------------------- utilities
__device__ __forceinline__ int iclamp(int v, int lo, int hi) {
    return v < lo ? lo : (v > hi ? hi : v);
}

// Gaussian 5-tap, sigma = 5/6 (k/6), normalized
__device__ __forceinline__ float gband(int t) {
    // g = {0.0269131, 0.2333660, 0.4794380, 0.2333660, 0.0269131}
    float r = 0.0f;
    r = (t == 0 || t == 4) ? 0.02691312f : r;
    r = (t == 1 || t == 3) ? 0.23336595f : r;
    r = (t == 2)           ? 0.47943792f : r;
    return r;
}

// Exact JAX threefry2x32 (20 rounds)
__device__ __forceinline__ void threefry2x32(uint32_t k0, uint32_t k1,
                                             uint32_t x0, uint32_t x1,
                                             uint32_t& o0, uint32_t& o1) {
    const uint32_t k2 = 0x1BD11BDAu ^ k0 ^ k1;
    x0 += k0; x1 += k1;
#define TFR(r) { x0 += x1; x1 = (x1 << (r)) | (x1 >> (32 - (r))); x1 ^= x0; }
    TFR(13) TFR(15) TFR(26) TFR(6)   x0 += k1; x1 += k2 + 1u;
    TFR(17) TFR(29) TFR(16) TFR(24)  x0 += k2; x1 += k0 + 2u;
    TFR(13) TFR(15) TFR(26) TFR(6)   x0 += k0; x1 += k1 + 3u;
    TFR(17) TFR(29) TFR(16) TFR(24)  x0 += k1; x1 += k2 + 4u;
    TFR(13) TFR(15) TFR(26) TFR(6)   x0 += k2; x1 += k0 + 5u;
#undef TFR
    o0 = x0; o1 = x1;
}

// uniform [0,1) for flat element index e of a length-2*halfn array (JAX semantics)
__device__ __forceinline__ float tf_uniform(size_t e, size_t halfn) {
    uint32_t c0, c1, o0, o1;
    bool hi = e >= halfn;
    if (hi) { c0 = (uint32_t)(e - halfn); c1 = (uint32_t)e; }
    else    { c0 = (uint32_t)e;           c1 = (uint32_t)(e + halfn); }
    threefry2x32(0u, 42u, c0, c1, o0, o1);
    uint32_t bits = hi ? o1 : o0;
    return __uint_as_float((bits >> 9) | 0x3F800000u) - 1.0f;
}

// ---------------------------------------------------------------- kernel 0: init min/max
__global__ void k_init_minmax(unsigned* __restrict__ mnb, unsigned* __restrict__ mxb) {
    int i = threadIdx.x;
    if (i < NB) { mnb[i] = 0x7F800000u; mxb[i] = 0u; }  // +inf / 0 (contrast >= 0)
}

// ---------------------------------------------------------------- kernel 1: contrast
// 32x32 output tile per block; 34x34 gray tile (halo 1) in LDS.
// RGB tiles are staged into LDS with CDNA5 async global->LDS copies.
__global__ __launch_bounds__(256) void k_contrast(const float* __restrict__ img,
                                                  float* __restrict__ contrast) {
    __shared__ float stage[3 * 34 * 35];
    __shared__ float gray[34 * 35];
    const int tid = threadIdx.x;
    const int bx = blockIdx.x * 32;
    const int by = blockIdx.y * 32;
    const int b  = blockIdx.z;
    const uint64_t base = (uint64_t)(uintptr_t)(img + (size_t)b * 3 * IMG_H * IMG_W);

    // ---- async stage: 3 channels x 34x34 (clamped addresses; OOB zeroed later)
    for (int i = tid; i < 3 * 34 * 34; i += 256) {
        int ch  = i / (34 * 34);
        int rem = i % (34 * 34);
        int r = rem / 34, c = rem % 34;
        int gy = iclamp(by + r - 1, 0, IMG_H - 1);
        int gx = iclamp(bx + c - 1, 0, IMG_W - 1);
        uint32_t goff  = (uint32_t)((((ch * IMG_H) + gy) * IMG_W + gx) * 4);
        uint32_t laddr = (uint32_t)(uintptr_t)&stage[(ch * 34 + r) * 35 + c];
        asm volatile("global_load_async_to_lds_b32 %0, %1, %2"
                     :: "v"(laddr), "v"(goff), "s"(base) : "memory");
    }
    asm volatile("s_wait_asynccnt 0" ::: "memory");
    __syncthreads();

    // ---- luma into LDS (zero outside image: matches zero-padded convs)
    for (int i = tid; i < 34 * 34; i += 256) {
        int r = i / 34, c = i % 34;
        int gy = by + r - 1, gx = bx + c - 1;
        float v = 0.0f;
        if (gy >= 0 && gy < IMG_H && gx >= 0 && gx < IMG_W) {
            float R = stage[(0 * 34 + r) * 35 + c];
            float G = stage[(1 * 34 + r) * 35 + c];
            float B = stage[(2 * 34 + r) * 35 + c];
            v = 0.299f * R + 0.587f * G + 0.114f * B;
        }
        gray[r * 35 + c] = v;
    }
    __syncthreads();

#define GL(rr, cc) gray[(rr) * 35 + (cc)]
    for (int i = tid; i < 32 * 32; i += 256) {
        int r = i / 32, c = i % 32;
        int y = by + r, x = bx + c;
        int lr = r + 1, lc = c + 1;
        float g00 = GL(lr-1, lc-1), g01 = GL(lr-1, lc), g02 = GL(lr-1, lc+1);
        float g10 = GL(lr,   lc-1), g11 = GL(lr,   lc), g12 = GL(lr,   lc+1);
        float g20 = GL(lr+1, lc-1), g21 = GL(lr+1, lc), g22 = GL(lr+1, lc+1);
        float gxv = -g00 + g02 - 2.0f * g10 + 2.0f * g12 - g20 + g22;
        float gyv = -g00 - 2.0f * g01 - g02 + g20 + 2.0f * g21 + g22;
        float sobel = sqrtf(gxv * gxv + gyv * gyv + 1e-8f);
        float lapl  = fabsf(4.0f * g11 - g01 - g10 - g12 - g21);
        // dx = gray[j+2]-gray[j], j = clamp(x-1,0,W-3)  (edge-padded central diff)
        int jx = iclamp(x - 1, 0, IMG_W - 3);
        float dxv = GL(lr, (jx + 2) - bx + 1) - GL(lr, jx - bx + 1);
        int jy = iclamp(y - 1, 0, IMG_H - 3);
        float dyv = GL((jy + 2) - by + 1, lc) - GL(jy - by + 1, lc);
        float gradm = sqrtf(dxv * dxv + dyv * dyv + 1e-8f);
        contrast[((size_t)b * IMG_H + y) * IMG_W + x] =
            0.5f * sobel + 0.3f * lapl + 0.2f * gradm;
    }
#undef GL
}

// ---------------------------------------------------------------- kernel 2: per-batch min/max
__global__ __launch_bounds__(256) void k_minmax(const float* __restrict__ contrast,
                                                unsigned* __restrict__ mnb,
                                                unsigned* __restrict__ mxb) {
    __shared__ float smin[256], smax[256];
    const int tid = threadIdx.x;
    const int b = blockIdx.y;
    const float* p = contrast + (size_t)b * IMG_H * IMG_W + (size_t)blockIdx.x * 4096;
    float lmin = __uint_as_float(0x7F800000u), lmax = 0.0f;
#pragma unroll
    for (int k = 0; k < 16; ++k) {
        float v = p[k * 256 + tid];
        lmin = fminf(lmin, v);
        lmax = fmaxf(lmax, v);
    }
    smin[tid] = lmin; smax[tid] = lmax;
    __syncthreads();
    for (int s = 128; s > 0; s >>= 1) {
        if (tid < s) {
            smin[tid] = fminf(smin[tid], smin[tid + s]);
            smax[tid] = fmaxf(smax[tid], smax[tid + s]);
        }
        __syncthreads();
    }
    if (tid == 0) {  // contrast >= 0 => uint order == float order
        atomicMin(&mnb[b], __float_as_uint(smin[0]));
        atomicMax(&mxb[b], __float_as_uint(smax[0]));
    }
}

// ---------------------------------------------------------------- kernel 3: density (WMMA blur)
// One wave per 16x16 tile. Horizontal 5-tap on VALU; vertical 5-tap as
// out(16x16) = V(16x20, banded) x T(20x16) via five chained v_wmma_f32_16x16x4_f32.
__global__ __launch_bounds__(256) void k_density(const float* __restrict__ contrast,
                                                 const unsigned* __restrict__ mnb,
                                                 const unsigned* __restrict__ mxb,
                                                 float* __restrict__ density) {
    __shared__ float Nsh[8][20 * 21];  // normalized input tile + halo2, stride 21
    __shared__ float Tsh[8][20 * 16];  // after horizontal pass
    const int tid  = threadIdx.x;
    const int w    = tid >> 5;
    const int lane = tid & 31;
    const int b = blockIdx.y;
    const int t = blockIdx.x * 8 + w;            // 0..4095
    const int ty0 = (t >> 6) * 16, tx0 = (t & 63) * 16;

    const float mn = __uint_as_float(mnb[b]);
    const float mx = __uint_as_float(mxb[b]);
    const bool  ok = mx > mn;
    const float inv = ok ? 1.0f / (mx - mn) : 0.0f;
    const float* p = contrast + (size_t)b * IMG_H * IMG_W;

    for (int i = lane; i < 400; i += 32) {       // 20x20 load+normalize (zero-pad)
        int r = i / 20, c = i % 20;
        int gy = ty0 + r - 2, gx = tx0 + c - 2;
        float v = 0.0f;
        if (gy >= 0 && gy < IMG_H && gx >= 0 && gx < IMG_W && ok)
            v = (p[(size_t)gy * IMG_W + gx] - mn) * inv;
        Nsh[w][r * 21 + c] = v;
    }
    __syncthreads();

    for (int i = lane; i < 320; i += 32) {       // horizontal taps -> T[20x16]
        int r = i / 16, x = i % 16;
        const float* row = &Nsh[w][r * 21];
        Tsh[w][r * 16 + x] = 0.02691312f * row[x]     + 0.23336595f * row[x + 1]
                           + 0.47943792f * row[x + 2] + 0.23336595f * row[x + 3]
                           + 0.02691312f * row[x + 4];
    }
    __syncthreads();

    const int half = lane >> 4, m = lane & 15;
    v8f acc = {};
#pragma unroll
    for (int k0 = 0; k0 < 20; k0 += 4) {
        int ka = k0 + half * 2;
        v2f a, bf;
        a.x  = gband(ka - m);                    // A[16x4]: V[m][ka], V[m][ka+1]
        a.y  = gband(ka + 1 - m);
        bf.x = Tsh[w][ka * 16 + m];              // B[4x16]:  T[ka][n],  T[ka+1][n]
        bf.y = Tsh[w][(ka + 1) * 16 + m];
        acc = __builtin_amdgcn_wmma_f32_16x16x4_f32(
            false, a, false, bf, (short)0, acc, false, false);
    }

    float* dp = density + (size_t)b * IMG_H * IMG_W;
#pragma unroll
    for (int i = 0; i < 8; ++i) {                // C/D layout: row i (lanes<16), i+8 (>=16)
        int y = ty0 + i + half * 8;
        int x = tx0 + m;
        float sm = acc[i];
        dp[(size_t)y * IMG_W + x] = 0.1f + 0.9f * sqrtf(fmaxf(sm, 0.0f));
    }
}

// ---------------------------------------------------------------- kernel 4: coords/weights
__global__ __launch_bounds__(256) void k_coords(const float* __restrict__ density,
                                                float* __restrict__ coords,
                                                float* __restrict__ weights) {
    const int gidx = blockIdx.x * 256 + threadIdx.x;     // [0, NB*TH*TW)
    const int tx = gidx % TW;
    const int ty = (gidx / TW) % TH;
    const int b  = gidx / (TW * TH);

    float d = density[((size_t)b * IMG_H + ty * 2) * IMG_W + tx * 2];
    int ns = d > 0.7f ? 4 : (d > 0.4f ? 2 : 1);
    float basey = -1.0f + 2.0f * (float)ty / 511.0f;
    float basex = -1.0f + 2.0f * (float)tx / 511.0f;
    const float cell = 2.0f / 512.0f;
    float useoff = (ns > 1) ? 1.0f : 0.0f;
    float invns = 1.0f / (float)ns;

    const size_t halfn = (size_t)NB * TH * TW * 8 / 2;   // 8,388,608
    const size_t eb = (size_t)gidx * 8;
    float* cp = coords  + (size_t)gidx * 8;
    float* wp = weights + (size_t)gidx * 4;

#pragma unroll
    for (int s = 0; s < 4; ++s) {
        float u0 = tf_uniform(eb + s * 2 + 0, halfn);
        float u1 = tf_uniform(eb + s * 2 + 1, halfn);
        cp[s * 2 + 0] = basey + useoff * (u0 - 0.5f) * 0.8f * cell;
        cp[s * 2 + 1] = basex + useoff * (u1 - 0.5f) * 0.8f * cell;
        wp[s] = (s < ns) ? d * invns : 0.0f;
    }
}

// ---------------------------------------------------------------- launch
extern "C" void kernel_launch(void* const* d_in, const int* in_sizes, int n_in,
                              void* d_out, int out_size, void* d_ws, size_t ws_size,
                              hipStream_t stream) {
    (void)in_sizes; (void)n_in; (void)out_size; (void)ws_size;
    const float* img = (const float*)d_in[0];
    // d_in[1]/d_in[2] = target_height/width (512/512, fixed by setup_inputs)

    float* out     = (float*)d_out;
    float* coords  = out;                                   // 8*512*512*4*2
    float* weights = out + (size_t)NB * TH * TW * 4 * 2;    // 8*512*512*4
    float* density = weights + (size_t)NB * TH * TW * 4;    // 8*1024*1024

    unsigned* mnb = (unsigned*)d_ws;
    unsigned* mxb = mnb + NB;
    float* contrast = (float*)((char*)d_ws + 256);          // 8*1024*1024 floats

    k_init_minmax<<<dim3(1), dim3(64), 0, stream>>>(mnb, mxb);
    k_contrast<<<dim3(32, 32, NB), dim3(256), 0, stream>>>(img, contrast);
    k_minmax<<<dim3(256, NB), dim3(256), 0, stream>>>(contrast, mnb, mxb);
    k_density<<<dim3(512, NB), dim3(256), 0, stream>>>(contrast, mnb, mxb, density);
    k_coords<<<dim3((NB * TH * TW) / 256), dim3(256), 0, stream>>>(density, coords, weights);
}